// kWTA_55035710931812
// MI455X (gfx1250) — compile-verified
//
#include <hip/hip_runtime.h>
#include <stdint.h>

#ifndef __has_builtin
#define __has_builtin(x) 0
#endif

#define ROWS 16384
#define COLS 4096
#define KSEL 205
#define TPB  256
#define VPT  (COLS / TPB)   // 16 values per thread

typedef int v4i __attribute__((ext_vector_type(4)));
typedef v4i __attribute__((address_space(1))) g_v4i;   // global (AS1)
typedef v4i __attribute__((address_space(3))) l_v4i;   // LDS (AS3)

// order-preserving map fp32 -> uint32 (ascending uint == ascending float)
__device__ __forceinline__ uint32_t f2ord(uint32_t b) {
    return (b & 0x80000000u) ? ~b : (b | 0x80000000u);
}
__device__ __forceinline__ uint32_t ord2f(uint32_t u) {
    return (u & 0x80000000u) ? (u & 0x7fffffffu) : ~u;
}

__global__ __launch_bounds__(TPB) void kwta_kernel(const float* __restrict__ x,
                                                   float* __restrict__ out) {
    __shared__ float    row[COLS];    // 16 KB staged row
    __shared__ uint32_t hist[256];    // radix histogram
    __shared__ uint32_t bcast[2];     // [0]=winning digit  [1]=count strictly above

    const int tid = threadIdx.x;
    const float* __restrict__ grow = x   + (size_t)blockIdx.x * COLS;
    float*       __restrict__ orow = out + (size_t)blockIdx.x * COLS;

    // ---- stage row into LDS via CDNA5 async global->LDS DMA (ASYNCcnt path) ----
#if __has_builtin(__builtin_amdgcn_global_load_async_to_lds_b128)
    #pragma unroll
    for (int i = 0; i < VPT / 4; ++i) {
        const int idx4 = i * TPB + tid;  // float4 index; lanes contiguous (512B/instr)
        g_v4i* gp = (g_v4i*)(uintptr_t)(grow + idx4 * 4);
        l_v4i* lp = (l_v4i*)(uint32_t)(uintptr_t)(&row[idx4 * 4]);
        __builtin_amdgcn_global_load_async_to_lds_b128(gp, lp, 0, 0);
    }
    hist[tid] = 0;   // pre-zero round-1 histogram while DMA is in flight
  #if __has_builtin(__builtin_amdgcn_s_wait_asynccnt)
    __builtin_amdgcn_s_wait_asynccnt(0);
  #else
    asm volatile("s_wait_asynccnt 0" ::: "memory");
  #endif
#else
    #pragma unroll
    for (int i = 0; i < VPT; ++i) row[tid + i * TPB] = grow[tid + i * TPB];
    hist[tid] = 0;
#endif
    __syncthreads();

    // ---- pull this thread's 16 keys into registers (conflict-free b32 reads) ----
    uint32_t u[VPT];
    #pragma unroll
    for (int i = 0; i < VPT; ++i)
        u[i] = f2ord(__float_as_uint(row[tid + i * TPB]));

    // ---- MSB-first radix select: exact key of the K-th largest element ----
    uint32_t prefix = 0, kmask = 0, krem = KSEL;
    #pragma unroll
    for (int r = 0; r < 4; ++r) {
        const int shift = 24 - 8 * r;

        // histogram of the surviving candidates (hist pre-zeroed)
        #pragma unroll
        for (int i = 0; i < VPT; ++i)
            if ((u[i] & kmask) == prefix)
                atomicAdd(&hist[(u[i] >> shift) & 0xFFu], 1u);
        __syncthreads();

        // single-wave suffix scan + winner-bin search (wave 0, 8 bins/lane)
        if (tid < 32) {
            uint32_t h[8], loc[8];
            #pragma unroll
            for (int k = 0; k < 8; ++k) h[k] = hist[tid * 8 + k];
            loc[7] = h[7];
            #pragma unroll
            for (int k = 6; k >= 0; --k) loc[k] = loc[k + 1] + h[k];

            // inclusive suffix sum of per-lane chunk totals across the wave
            uint32_t s = loc[0];
            #pragma unroll
            for (int off = 1; off < 32; off <<= 1) {
                uint32_t t = __shfl_down(s, off);
                if (tid + off < 32) s += t;
            }
            uint32_t above_chunk = __shfl_down(s, 1);   // suffix of lanes > tid
            if (tid == 31) above_chunk = 0;

            #pragma unroll
            for (int k = 0; k < 8; ++k) {
                uint32_t here = loc[k] + above_chunk;                       // #elems with digit >= d
                uint32_t nxt  = (k == 7) ? above_chunk : loc[k + 1] + above_chunk; // digit > d
                if (here >= krem && nxt < krem) {        // unique winner bin
                    bcast[0] = (uint32_t)(tid * 8 + k);
                    bcast[1] = nxt;
                }
            }
        }
        __syncthreads();

        prefix |= bcast[0] << shift;
        krem   -= bcast[1];
        kmask  |= 0xFFu << shift;

        if (r < 3) {                    // re-zero for next round
            hist[tid] = 0;
            __syncthreads();
        }
    }

    // ---- write winners, zero the rest (streaming stores) ----
    const uint32_t thr = prefix;  // exact K-th largest key
    #pragma unroll
    for (int i = 0; i < VPT; ++i) {
        float v = (u[i] >= thr) ? __uint_as_float(ord2f(u[i])) : 0.0f;
        __builtin_nontemporal_store(v, &orow[tid + i * TPB]);
    }
}

extern "C" void kernel_launch(void* const* d_in, const int* in_sizes, int n_in,
                              void* d_out, int out_size, void* d_ws, size_t ws_size,
                              hipStream_t stream) {
    const float* x   = (const float*)d_in[0];
    float*       out = (float*)d_out;
    (void)in_sizes; (void)n_in; (void)out_size; (void)d_ws; (void)ws_size;
    kwta_kernel<<<dim3(ROWS), dim3(TPB), 0, stream>>>(x, out);
}